// GCN_E_2_4209067950533
// MI455X (gfx1250) — compile-verified
//
#include <hip/hip_runtime.h>
#include <hip/hip_bf16.h>

typedef __attribute__((ext_vector_type(2))) float v2f;
typedef __attribute__((ext_vector_type(8))) float v8f;

// ---------------------------------------------------------------------------
// Dense GEMM  C[M,N] = A[M,K] * B[K,N]  via V_WMMA_F32_16X16X4_F32 (wave32).
// Each wave computes a 16-row x 64-col slab (4 x 16x16 accumulators).
// Block = 256 threads = 8 waves; waves grouped so all waves sharing an A strip
// live in one block (A strip read once per WGP, B stays L2-resident).
// ---------------------------------------------------------------------------
template <int K, int N>
__global__ __launch_bounds__(256) void gemm_wmma_f32(
    const float* __restrict__ A, const float* __restrict__ B,
    float* __restrict__ C, int M) {
  constexpr int NT = 4;                          // 16-col tiles per wave
  constexpr int WAVES_PER_STRIP = N / (16 * NT); // 256 -> 4, 128 -> 2
  constexpr int STRIPS_PER_BLOCK = 8 / WAVES_PER_STRIP;

  const int wave = threadIdx.x >> 5;
  const int lane = threadIdx.x & 31;
  const int strip = blockIdx.x * STRIPS_PER_BLOCK + wave / WAVES_PER_STRIP;
  if (strip * 16 >= M) return;                   // uniform per wave: EXEC stays full
  const int m0 = strip * 16;
  const int n0 = (wave % WAVES_PER_STRIP) * (16 * NT);

  const int lr   = lane & 15;        // row (A) / col (B,C) within tile
  const int khalf = lane >> 4;       // 0: K=0,1   1: K=2,3
  const int koff = khalf << 1;

  const float* aptr = A + (long long)(m0 + lr) * K + koff;
  const float* bptr = B + (long long)koff * N + n0 + lr;

  v8f acc[NT];
#pragma unroll
  for (int t = 0; t < NT; ++t) acc[t] = (v8f){0.f,0.f,0.f,0.f,0.f,0.f,0.f,0.f};

  for (int k = 0; k < K; k += 4) {
    v2f a;
    a.x = aptr[0];                   // contiguous pair -> global_load_b64
    a.y = aptr[1];
    aptr += 4;
#pragma unroll
    for (int t = 0; t < NT; ++t) {
      v2f b;
      b.x = bptr[t * 16];            // B[k+koff][nt+lr]
      b.y = bptr[t * 16 + N];        // B[k+koff+1][nt+lr]
      acc[t] = __builtin_amdgcn_wmma_f32_16x16x4_f32(
          /*neg_a=*/false, a, /*neg_b=*/false, b,
          /*c_mod=*/(short)0, acc[t], /*reuse_a=*/false, /*reuse_b=*/false);
    }
    bptr += 4 * N;
  }

  // C/D layout: VGPR r -> row m0 + r + 8*khalf, col = nt + lr
  const int rbase = m0 + (khalf << 3);
#pragma unroll
  for (int t = 0; t < NT; ++t) {
#pragma unroll
    for (int r = 0; r < 8; ++r) {
      C[(long long)(rbase + r) * N + n0 + t * 16 + lr] = acc[t][r];
    }
  }
}

// ---------------------------------------------------------------------------
// Zero-fill fp32 buffer (grid-stride).
// ---------------------------------------------------------------------------
__global__ __launch_bounds__(256) void zero_f32(float* __restrict__ p, long long n) {
  long long i = (long long)blockIdx.x * blockDim.x + threadIdx.x;
  long long stride = (long long)gridDim.x * blockDim.x;
  for (; i < n; i += stride) p[i] = 0.0f;
}

// ---------------------------------------------------------------------------
// COO SpMM scatter: ACC[row[e], :] += w[e] * S[col[e], :]
// Thread i handles (edge = i >> log2(D), dim = i & (D-1)): S reads and the
// atomics are fully coalesced; accumulator fits in L2 so atomics resolve there.
// ---------------------------------------------------------------------------
template <int D, int LOG2D>
__global__ __launch_bounds__(256) void spmm_scatter(
    const int* __restrict__ rows, const int* __restrict__ cols,
    const float* __restrict__ w, const float* __restrict__ S,
    float* __restrict__ ACC, int nE) {
  long long total = (long long)nE << LOG2D;
  long long i = (long long)blockIdx.x * blockDim.x + threadIdx.x;
  long long stride = (long long)gridDim.x * blockDim.x;
  for (; i < total; i += stride) {
    int e = (int)(i >> LOG2D);
    int d = (int)(i & (D - 1));
    float v = w[e] * S[(long long)cols[e] * D + d];
    atomicAdd(&ACC[(long long)rows[e] * D + d], v);
  }
}

// ---------------------------------------------------------------------------
// out = leaky_relu(acc + bias), elementwise (in-place safe).
// ---------------------------------------------------------------------------
template <int D, int LOG2D>
__global__ __launch_bounds__(256) void bias_leaky(
    const float* __restrict__ acc, const float* __restrict__ bias,
    float* __restrict__ out, long long total) {
  long long i = (long long)blockIdx.x * blockDim.x + threadIdx.x;
  long long stride = (long long)gridDim.x * blockDim.x;
  for (; i < total; i += stride) {
    int d = (int)(i & (D - 1));
    float h = acc[i] + bias[d];
    out[i] = h > 0.0f ? h : 0.25f * h;
  }
}

// ---------------------------------------------------------------------------
// Host-side orchestration.
// Inputs: x, W1, b1, W2, b2, adj_row, adj_col, edge_weight
// ---------------------------------------------------------------------------
extern "C" void kernel_launch(void* const* d_in, const int* in_sizes, int n_in,
                              void* d_out, int out_size, void* d_ws, size_t ws_size,
                              hipStream_t stream) {
  constexpr int NN = 100000, IN = 512, HD = 256, OD = 128, NE = 3200000;

  const float* x   = (const float*)d_in[0];
  const float* W1  = (const float*)d_in[1];
  const float* b1  = (const float*)d_in[2];
  const float* W2  = (const float*)d_in[3];
  const float* b2  = (const float*)d_in[4];
  const int*   arw = (const int*)d_in[5];
  const int*   acl = (const int*)d_in[6];
  const float* ew  = (const float*)d_in[7];
  float* out = (float*)d_out;

  // Workspace layout: [ S1 : NN*HD ][ A1 : NN*HD ]; S2 reuses S1's slot.
  float* S1 = (float*)d_ws;
  float* A1 = S1 + (size_t)NN * HD;
  float* S2 = S1;

  const int STRIPS = NN / 16;  // 6250

  // ---- Layer 1 ----
  // S1 = x @ W1
  gemm_wmma_f32<IN, HD><<<STRIPS / 2, 256, 0, stream>>>(x, W1, S1, NN);

  // A1 = 0
  {
    long long n = (long long)NN * HD;
    int grid = (int)((n + 256 * 8 - 1) / (256 * 8));
    zero_f32<<<grid, 256, 0, stream>>>(A1, n);
  }

  // A1[r] += w * S1[c]  (COO scatter)
  spmm_scatter<HD, 8><<<131072, 256, 0, stream>>>(arw, acl, ew, S1, A1, NE);

  // A1 = leaky(A1 + b1)   (in place)
  bias_leaky<HD, 8><<<65536, 256, 0, stream>>>(A1, b1, A1, (long long)NN * HD);

  // ---- Layer 2 ----
  // S2 = A1 @ W2
  gemm_wmma_f32<HD, OD><<<(STRIPS + 3) / 4, 256, 0, stream>>>(A1, W2, S2, NN);

  // out = 0
  {
    long long n = (long long)NN * OD;
    int grid = (int)((n + 256 * 8 - 1) / (256 * 8));
    zero_f32<<<grid, 256, 0, stream>>>(out, n);
  }

  // out[r] += w * S2[c]
  spmm_scatter<OD, 7><<<131072, 256, 0, stream>>>(arw, acl, ew, S2, out, NE);

  // out = leaky(out + b2)  (in place)
  bias_leaky<OD, 7><<<65536, 256, 0, stream>>>(out, b2, out, (long long)NN * OD);
}